// EnergyFunction_40604620816440
// MI455X (gfx1250) — compile-verified
//
#include <hip/hip_runtime.h>
#include <math.h>

// Problem constants (match reference)
#define BQ   2048      // batch rows
#define NQ   65536     // codebook rows
#define DQ   256       // feature dim
#define KQ   32        // top-k
#define BINS 512       // histogram bins over sims in [-1, 1]
#define CAP  192       // candidate capacity per row (expected ~36 pass threshold)
static constexpr float TEMP_ = 0.1f;
static constexpr float EPS_  = 1e-4f;

typedef float v2f __attribute__((ext_vector_type(2)));
typedef float v8f __attribute__((ext_vector_type(8)));

// FP32 WMMA: D(16x16,f32) = A(16x4,f32) * B(4x16,f32) + C
// 8-arg pattern per probe doc: (neg_a, A, neg_b, B, c_mod, C, reuse_a, reuse_b)
__device__ __forceinline__ v8f wmma_f32(v2f a, v2f b, v8f c) {
  return __builtin_amdgcn_wmma_f32_16x16x4_f32(false, a, false, b, (short)0, c,
                                               false, false);
}

// ---------------------------------------------------------------------------
// zero workspace words
__global__ void k_zero(unsigned* __restrict__ p, int nwords) {
  int i = blockIdx.x * blockDim.x + threadIdx.x;
  if (i < nwords) p[i] = 0u;
}

// ---------------------------------------------------------------------------
// Pass 1: sims = x @ mu^T via FP32 WMMA, per-row histogram of values.
// grid = (16 col-chunks of 4096, 128 row-tiles of 16), block = 256 (8 waves)
__global__ __launch_bounds__(256)
void k_hist(const float* __restrict__ x, const float* __restrict__ mu,
            unsigned* __restrict__ ghist) {
  __shared__ float    xs[16 * DQ];        // 16 KB: x tile, reused by all waves
  __shared__ unsigned hist[16 * BINS];    // 32 KB: per-row histograms

  const int tid     = threadIdx.x;
  const int rowbase = blockIdx.y * 16;
  const int colchunk = blockIdx.x * 4096;

  for (int i = tid; i < 16 * BINS; i += 256) hist[i] = 0u;
  for (int i = tid; i < 16 * DQ / 4; i += 256)
    ((float4*)xs)[i] = ((const float4*)(x + (size_t)rowbase * DQ))[i];
  __syncthreads();

  const int w = tid >> 5, lane = tid & 31;
  const int rsel = lane & 15;        // M for A-lanes / N for B-lanes / N for C
  const int half = lane >> 4;        // selects K pair {0,1} vs {2,3}
  const int ksel = half * 2;

  for (int t = w; t < 256; t += 8) { // 256 column tiles per chunk, 8 waves
    const int col0 = colchunk + t * 16;
    const float* aptr = xs + rsel * DQ + ksel;
    const float* bptr = mu + (size_t)(col0 + rsel) * DQ + ksel;
    v8f acc = {0.f, 0.f, 0.f, 0.f, 0.f, 0.f, 0.f, 0.f};
#pragma unroll 8
    for (int kk = 0; kk < DQ; kk += 4) {
      v2f a = *(const v2f*)(aptr + kk);
      v2f b = *(const v2f*)(bptr + kk);
      acc = wmma_f32(a, b, acc);
    }
#pragma unroll
    for (int r = 0; r < 8; ++r) {
      float val = acc[r];
      int rl = r + half * 8;  // local row 0..15 (C layout)
      int bin = (int)floorf((val + 1.0f) * (float)(BINS / 2));
      bin = bin < 0 ? 0 : (bin > BINS - 1 ? BINS - 1 : bin);
      atomicAdd(&hist[rl * BINS + bin], 1u);
    }
  }
  __syncthreads();
  for (int i = tid; i < 16 * BINS; i += 256) {
    unsigned c = hist[i];
    if (c)
      atomicAdd(&ghist[(size_t)(rowbase + (i / BINS)) * BINS + (i % BINS)], c);
  }
}

// ---------------------------------------------------------------------------
// Per-row threshold = lower edge of bin containing the 32nd largest value.
__global__ void k_thresh(const unsigned* __restrict__ ghist,
                         float* __restrict__ thresh) {
  int row = blockIdx.x * blockDim.x + threadIdx.x;
  if (row >= BQ) return;
  const unsigned* h = ghist + (size_t)row * BINS;
  unsigned cum = 0;
  int sel = 0;
  for (int b = BINS - 1; b >= 0; --b) {
    cum += h[b];
    if (cum >= KQ) { sel = b; break; }
  }
  thresh[row] = (float)sel * (2.0f / BINS) - 1.0f;
}

// ---------------------------------------------------------------------------
// Pass 2: recompute sims (bitwise identical), collect candidates >= threshold.
__global__ __launch_bounds__(256)
void k_collect(const float* __restrict__ x, const float* __restrict__ mu,
               const float* __restrict__ thresh, unsigned* __restrict__ cnt,
               float* __restrict__ cval, int* __restrict__ cidx) {
  __shared__ float xs[16 * DQ];
  __shared__ float thr[16];

  const int tid     = threadIdx.x;
  const int rowbase = blockIdx.y * 16;
  const int colchunk = blockIdx.x * 4096;

  if (tid < 16) thr[tid] = thresh[rowbase + tid];
  for (int i = tid; i < 16 * DQ / 4; i += 256)
    ((float4*)xs)[i] = ((const float4*)(x + (size_t)rowbase * DQ))[i];
  __syncthreads();

  const int w = tid >> 5, lane = tid & 31;
  const int rsel = lane & 15;
  const int half = lane >> 4;
  const int ksel = half * 2;

  for (int t = w; t < 256; t += 8) {
    const int col0 = colchunk + t * 16;
    const float* aptr = xs + rsel * DQ + ksel;
    const float* bptr = mu + (size_t)(col0 + rsel) * DQ + ksel;
    v8f acc = {0.f, 0.f, 0.f, 0.f, 0.f, 0.f, 0.f, 0.f};
#pragma unroll 8
    for (int kk = 0; kk < DQ; kk += 4) {
      v2f a = *(const v2f*)(aptr + kk);
      v2f b = *(const v2f*)(bptr + kk);
      acc = wmma_f32(a, b, acc);
    }
#pragma unroll
    for (int r = 0; r < 8; ++r) {
      float val = acc[r];
      int rl = r + half * 8;
      if (val >= thr[rl]) {
        int row = rowbase + rl;
        int col = col0 + rsel;  // C layout: N = lane & 15
        unsigned p = atomicAdd(&cnt[row], 1u);
        if (p < CAP) {
          cval[(size_t)row * CAP + p] = val;
          cidx[(size_t)row * CAP + p] = col;
        }
      }
    }
  }
}

// ---------------------------------------------------------------------------
// e_geom partials: bs = x @ x^T, sum off-diagonal -log(1 - bs + eps).
// grid = 128 (row-tiles), block = 256; deterministic per-block partial.
__global__ __launch_bounds__(256)
void k_geom(const float* __restrict__ x, float* __restrict__ partial) {
  __shared__ float xs[16 * DQ];
  __shared__ float red[256];

  const int tid     = threadIdx.x;
  const int rowbase = blockIdx.x * 16;

  for (int i = tid; i < 16 * DQ / 4; i += 256)
    ((float4*)xs)[i] = ((const float4*)(x + (size_t)rowbase * DQ))[i];
  __syncthreads();

  const int w = tid >> 5, lane = tid & 31;
  const int rsel = lane & 15;
  const int half = lane >> 4;
  const int ksel = half * 2;

  float s = 0.0f;
  for (int t = w; t < 128; t += 8) {  // 128 column tiles cover all of B
    const int col0 = t * 16;
    const float* aptr = xs + rsel * DQ + ksel;
    const float* bptr = x + (size_t)(col0 + rsel) * DQ + ksel;
    v8f acc = {0.f, 0.f, 0.f, 0.f, 0.f, 0.f, 0.f, 0.f};
#pragma unroll 8
    for (int kk = 0; kk < DQ; kk += 4) {
      v2f a = *(const v2f*)(aptr + kk);
      v2f b = *(const v2f*)(bptr + kk);
      acc = wmma_f32(a, b, acc);
    }
#pragma unroll
    for (int r = 0; r < 8; ++r) {
      int row = rowbase + r + half * 8;
      int col = col0 + rsel;
      if (row != col) s += -logf(1.0f - acc[r] + EPS_);
    }
  }
  red[tid] = s;
  __syncthreads();
  for (int off = 128; off > 0; off >>= 1) {
    if (tid < off) red[tid] += red[tid + off];
    __syncthreads();
  }
  if (tid == 0) partial[blockIdx.x] = red[0];
}

__global__ void k_reduce_geom(const float* __restrict__ partial,
                              float* __restrict__ geom) {
  __shared__ float red[128];
  int tid = threadIdx.x;
  red[tid] = partial[tid];
  __syncthreads();
  for (int off = 64; off > 0; off >>= 1) {
    if (tid < off) red[tid] += red[tid + off];
    __syncthreads();
  }
  if (tid == 0) geom[0] = red[0] / ((float)BQ * (float)(BQ - 1));
}

// ---------------------------------------------------------------------------
// Exact top-32 from candidates (ties: val desc, idx asc — matches lax.top_k),
// then e_splat (stable logsumexp), e_comp, and final combine.
__global__ void k_final(const unsigned* __restrict__ cnt,
                        const float* __restrict__ cval,
                        const int* __restrict__ cidx,
                        const float* __restrict__ alpha,
                        const float* __restrict__ W,
                        const float* __restrict__ bias,
                        const float* __restrict__ geom,
                        float* __restrict__ out) {
  int row = blockIdx.x * blockDim.x + threadIdx.x;
  if (row >= BQ) return;
  unsigned c = cnt[row];
  int n = (c < (unsigned)CAP) ? (int)c : CAP;
  const float* cv = cval + (size_t)row * CAP;
  const int*   ci = cidx + (size_t)row * CAP;

  float lastV = 3.0e38f;
  int   lastI = -1;
  float ex[KQ];
  float u = 0.0f, v = 0.0f;
  float m = -3.0e38f;

  for (int j = 0; j < KQ; ++j) {
    float bestV = -3.0e38f;
    int   bestI = 0x7fffffff;
    for (int i = 0; i < n; ++i) {
      float vv = cv[i];
      int   ii = ci[i];
      // must rank strictly after the previously selected (val desc, idx asc)
      bool after = (vv < lastV) || (vv == lastV && ii > lastI);
      if (!after) continue;
      bool better = (vv > bestV) || (vv == bestV && ii < bestI);
      if (better) { bestV = vv; bestI = ii; }
    }
    float val = (bestI == 0x7fffffff) ? -1.0f : bestV;  // fallback, unreachable
    int   idx = (bestI == 0x7fffffff) ? 0 : bestI;
    lastV = val;
    lastI = idx;
    if (j == 0) u = val;
    if (j == 1) v = val;
    float e = alpha[idx] * (val - 1.0f) * (1.0f / TEMP_);
    ex[j] = e;
    m = fmaxf(m, e);
  }
  float ssum = 0.0f;
  for (int j = 0; j < KQ; ++j) ssum += expf(ex[j] - m);
  float lse = m + logf(ssum);

  float z   = W[0] * u + W[1] * v + W[2] * (u * v) + bias[0];
  float sig = 1.0f / (1.0f + expf(-z));

  out[row] = -lse + 0.01f * geom[0] + 0.05f * sig;
}

// ---------------------------------------------------------------------------
extern "C" void kernel_launch(void* const* d_in, const int* in_sizes, int n_in,
                              void* d_out, int out_size, void* d_ws,
                              size_t ws_size, hipStream_t stream) {
  (void)in_sizes; (void)n_in; (void)out_size; (void)ws_size;
  const float* x     = (const float*)d_in[0];  // (2048, 256)
  const float* mu    = (const float*)d_in[1];  // (65536, 256)
  const float* alpha = (const float*)d_in[2];  // (65536,)
  const float* W     = (const float*)d_in[3];  // (1, 3)
  const float* bias  = (const float*)d_in[4];  // (1,)
  float* out = (float*)d_out;                  // (2048,)

  // Workspace layout (zero-needed regions first, contiguous)
  char* ws = (char*)d_ws;
  size_t off = 0;
  unsigned* ghist = (unsigned*)(ws + off); off += (size_t)BQ * BINS * 4;  // 4 MB
  unsigned* cnt   = (unsigned*)(ws + off); off += (size_t)BQ * 4;
  float*    partial = (float*)(ws + off);  off += 128 * 4;
  float*    geom  = (float*)(ws + off);    off += 4;
  float*    thresh = (float*)(ws + off);   off += (size_t)BQ * 4;
  float*    cval  = (float*)(ws + off);    off += (size_t)BQ * CAP * 4;
  int*      cidx  = (int*)(ws + off);      off += (size_t)BQ * CAP * 4;

  int zero_words = BQ * BINS + BQ + 128 + 1;
  k_zero<<<(zero_words + 255) / 256, 256, 0, stream>>>(ghist, zero_words);

  dim3 ggrid(16, 128);  // 16 col-chunks x 128 row-tiles
  k_hist<<<ggrid, 256, 0, stream>>>(x, mu, ghist);
  k_thresh<<<BQ / 256, 256, 0, stream>>>(ghist, thresh);
  k_collect<<<ggrid, 256, 0, stream>>>(x, mu, thresh, cnt, cval, cidx);
  k_geom<<<128, 256, 0, stream>>>(x, partial);
  k_reduce_geom<<<1, 128, 0, stream>>>(partial, geom);
  k_final<<<BQ / 256, 256, 0, stream>>>(cnt, cval, cidx, alpha, W, bias, geom,
                                        out);
}